// TVGNN_43155831390848
// MI455X (gfx1250) — compile-verified
//
#include <hip/hip_runtime.h>
#include <math.h>
#include <stdint.h>

#define TV_DELTA 1.6f
#define TV_EPS   1e-3f
#define BATCH    8

typedef __attribute__((ext_vector_type(2))) float v2f;
typedef __attribute__((ext_vector_type(8))) float v8f;

// ---------------------------------------------------------------------------
// fp32 WMMA helpers: one wave32 computes a 16x16 f32 C tile, K-loop of
// V_WMMA_F32_16X16X4_F32.  Layouts per CDNA5 ISA 7.12.2:
//   A 16x4 : v0 = K{0,2}, v1 = K{1,3}; lanes 0-15 -> Kpair{0,1}, 16-31 -> {2,3}
//   B 4x16 : same striping with lane = column
//   C 16x16: vgpr r holds row r (lanes 0-15) and row r+8 (lanes 16-31)
// ---------------------------------------------------------------------------
__device__ __forceinline__ v8f wmma_acc(v2f a, v2f b, v8f c) {
  return __builtin_amdgcn_wmma_f32_16x16x4_f32(false, a, false, b, (short)0, c,
                                               false, false);
}

// C += A(16xK, row-major, lda) * B(Kx16, row-major, ldb)
__device__ inline v8f gemm_tile_nn(const float* Ap, int lda,
                                   const float* Bp, int ldb, int K, v8f c) {
  const int lane = threadIdx.x & 31;
  const int m  = lane & 15;
  const int kh = (lane >> 4) << 1;  // 0 or 2
  for (int k0 = 0; k0 < K; k0 += 4) {
    v2f a, b;
    a.x = Ap[m * lda + k0 + kh];
    a.y = Ap[m * lda + k0 + kh + 1];
    b.x = Bp[(k0 + kh) * ldb + m];
    b.y = Bp[(k0 + kh + 1) * ldb + m];
    c = wmma_acc(a, b, c);
  }
  return c;
}

// C += A^T * B where logical A(m,k) = Ap[k*lda + m]  (for S^T x, S^T A S)
__device__ inline v8f gemm_tile_tn(const float* Ap, int lda,
                                   const float* Bp, int ldb, int K, v8f c) {
  const int lane = threadIdx.x & 31;
  const int m  = lane & 15;
  const int kh = (lane >> 4) << 1;
  for (int k0 = 0; k0 < K; k0 += 4) {
    v2f a, b;
    a.x = Ap[(k0 + kh) * lda + m];
    a.y = Ap[(k0 + kh + 1) * lda + m];
    b.x = Bp[(k0 + kh) * ldb + m];
    b.y = Bp[(k0 + kh + 1) * ldb + m];
    c = wmma_acc(a, b, c);
  }
  return c;
}

__device__ inline void store_tile(float* Dp, int ldd, v8f c) {
  const int lane = threadIdx.x & 31;
  const int n    = lane & 15;
  const int half = lane >> 4;
#pragma unroll
  for (int r = 0; r < 8; ++r) Dp[(r + 8 * half) * ldd + n] = c[r];
}

__device__ __forceinline__ float squareplus(float v) {
  return 0.5f * (sqrtf(v * v + 4.0f) + v);
}

// ---------------------------------------------------------------------------
// Async bulk staging: global -> LDS via GLOBAL_LOAD_ASYNC_TO_LDS_B128
// (ASYNCcnt-tracked DMA into LDS, GV addressing: VDST = 32-bit LDS byte
//  offset, VADDR = 64-bit global address).  nfloats must be a multiple of 4;
// both pointers 16B aligned.  Waits on s_wait_asynccnt itself; caller then does a barrier.
// ---------------------------------------------------------------------------
__device__ __forceinline__ void stage_to_lds_async(float* lds_dst,
                                                   const float* gsrc,
                                                   int nfloats) {
  uint32_t lbase =
      (uint32_t)(size_t)(__attribute__((address_space(3))) char*)lds_dst;
  const int nvec = nfloats >> 2;  // 16-byte chunks
  for (int t = threadIdx.x; t < nvec; t += blockDim.x) {
    uint32_t loff = lbase + 16u * (uint32_t)t;
    const float* g = gsrc + 4 * t;
    asm volatile("global_load_async_to_lds_b128 %0, %1, off"
                 :: "v"(loff), "v"(g)
                 : "memory");
  }
  asm volatile("s_wait_asynccnt 0x0" ::: "memory");
}

// ---------------------------------------------------------------------------
// xw[b] = X[b] @ W        grid: BATCH*(N/16)*(NO/16) waves of 32
// ---------------------------------------------------------------------------
__global__ void k_gemm_xw(const float* __restrict__ X, const float* __restrict__ W,
                          float* __restrict__ Y, int N, int K, int NO) {
  int tile = blockIdx.x;
  int nf = NO >> 4;
  int f0 = (tile % nf) << 4; tile /= nf;
  int nr = N >> 4;
  int i0 = (tile % nr) << 4;
  int b  = tile / nr;
  v8f c = {};
  c = gemm_tile_nn(X + ((size_t)b * N + i0) * K, K, W + f0, NO, K, c);
  store_tile(Y + ((size_t)b * N + i0) * NO + f0, NO, c);
}

// ---------------------------------------------------------------------------
// gamma[b,i,j] = a/max(L1(x_i,x_j), eps);  deg[b,i] = row sum.
// One workgroup per (16-row tile, b); xw[b] staged in LDS via async DMA.
// ---------------------------------------------------------------------------
__global__ void k_gamma(const float* __restrict__ xw, const float* __restrict__ A,
                        float* __restrict__ G, float* __restrict__ deg, int N) {
  extern __shared__ float sm[];
  float* sx  = sm;            // N*32
  float* red = sm + N * 32;   // 256
  const int b  = blockIdx.y;
  const int i0 = blockIdx.x * 16;
  stage_to_lds_async(sx, xw + (size_t)b * N * 32, N * 32);
  __syncthreads();
  const int il = threadIdx.x >> 4;
  const int jl = threadIdx.x & 15;
  const int i  = i0 + il;
  const float* xi   = sx + i * 32;
  const float* arow = A + ((size_t)b * N + i) * N;
  float* grow       = G + ((size_t)b * N + i) * N;
  float acc = 0.f;
  for (int j = jl; j < N; j += 16) {
    const float* xj = sx + j * 32;
    float d = 0.f;
#pragma unroll
    for (int f = 0; f < 32; ++f) d += fabsf(xi[f] - xj[f]);
    float g = arow[j] / fmaxf(d, TV_EPS);
    grow[j] = g;
    acc += g;
  }
  red[threadIdx.x] = acc;
  __syncthreads();
  if (jl == 0) {
    float s = 0.f;
    for (int t = 0; t < 16; ++t) s += red[il * 16 + t];
    deg[(size_t)b * N + i] = s;
  }
}

// ---------------------------------------------------------------------------
// Y = squareplus( xw - DELTA*(deg*xw - gamma@xw) + bias ).  gamma@xw via WMMA.
// grid: BATCH*(N/16)*2 waves.
// ---------------------------------------------------------------------------
__global__ void k_conv_out(const float* __restrict__ xw, const float* __restrict__ G,
                           const float* __restrict__ deg, const float* __restrict__ bias,
                           float* __restrict__ Y, int N) {
  int tile = blockIdx.x;
  int f0 = (tile & 1) << 4; tile >>= 1;
  int nr = N >> 4;
  int i0 = (tile % nr) << 4;
  int b  = tile / nr;
  v8f c = {};
  c = gemm_tile_nn(G + ((size_t)b * N + i0) * N, N,
                   xw + (size_t)b * N * 32 + f0, 32, N, c);
  const int lane = threadIdx.x & 31;
  const int n    = lane & 15;
  const int half = lane >> 4;
#pragma unroll
  for (int r = 0; r < 8; ++r) {
    int i = i0 + r + 8 * half;
    int f = f0 + n;
    float xv = xw[((size_t)b * N + i) * 32 + f];
    float v  = xv - TV_DELTA * (deg[(size_t)b * N + i] * xv - c[r]) + bias[f];
    Y[((size_t)b * N + i) * 32 + f] = squareplus(v);
  }
}

// ---------------------------------------------------------------------------
// WMMA dense with fused bias + squareplus: rows, IN, OUT all multiples of 16
// (and IN % 4 == 0).  Used for level-0 / level-2 pooling MLPs.
// grid: (rows/16)*(OUT/16) waves of 32.
// ---------------------------------------------------------------------------
__global__ void k_dense_wmma(const float* __restrict__ X, const float* __restrict__ W,
                             const float* __restrict__ bias, float* __restrict__ Y,
                             int rows, int IN, int OUT, int act) {
  int tile = blockIdx.x;
  int nf = OUT >> 4;
  int f0 = (tile % nf) << 4;
  int r0 = (tile / nf) << 4;
  v8f c = {};
  c = gemm_tile_nn(X + (size_t)r0 * IN, IN, W + f0, OUT, IN, c);
  const int lane = threadIdx.x & 31;
  const int n    = lane & 15;
  const int half = lane >> 4;
  float bv = bias[f0 + n];
#pragma unroll
  for (int r = 0; r < 8; ++r) {
    float v = c[r] + bv;
    if (act) v = squareplus(v);
    Y[(size_t)(r0 + r + 8 * half) * OUT + f0 + n] = v;
  }
}

// ---------------------------------------------------------------------------
// Plain dense (level-1 MLP with dims 42/53 that don't tile to 16).
// ---------------------------------------------------------------------------
__global__ void k_dense(const float* __restrict__ X, const float* __restrict__ W,
                        const float* __restrict__ bias, float* __restrict__ Y,
                        int rows, int IN, int OUT, int act) {
  int idx = blockIdx.x * blockDim.x + threadIdx.x;
  if (idx >= rows * OUT) return;
  int o = idx % OUT;
  int r = idx / OUT;
  const float* xr = X + (size_t)r * IN;
  __builtin_prefetch(xr, 0, 3);
  __builtin_prefetch(W + o, 0, 3);
  float s = bias[o];
  for (int i = 0; i < IN; ++i) s += xr[i] * W[i * OUT + o];
  if (act) s = squareplus(s);
  Y[idx] = s;
}

__global__ void k_softmax(const float* __restrict__ L, float* __restrict__ S,
                          int rows, int K) {
  int r = blockIdx.x * blockDim.x + threadIdx.x;
  if (r >= rows) return;
  const float* lr = L + (size_t)r * K;
  float m = -3.4e38f;
  for (int k = 0; k < K; ++k) m = fmaxf(m, lr[k]);
  float sum = 0.f;
  for (int k = 0; k < K; ++k) sum += expf(lr[k] - m);
  float inv = 1.f / sum;
  float* sr = S + (size_t)r * K;
  for (int k = 0; k < K; ++k) sr[k] = expf(lr[k] - m) * inv;
}

// ---------------------------------------------------------------------------
// TV loss partials: per (tile,b): sum_ij a_ij * L1(s_i,s_j) and nnz(a).
// s[b] staged in LDS via async DMA (max 128 KB < 320 KB/WGP).
// ---------------------------------------------------------------------------
__global__ void k_tv(const float* __restrict__ S, const float* __restrict__ A,
                     float* __restrict__ part, int N, int K) {
  extern __shared__ float sm[];
  float* ss  = sm;             // N*K
  float* red = sm + N * K;     // 512
  const int b  = blockIdx.y;
  const int i0 = blockIdx.x * 16;
  stage_to_lds_async(ss, S + (size_t)b * N * K, N * K);
  __syncthreads();
  const int il = threadIdx.x >> 4;
  const int jl = threadIdx.x & 15;
  const int i  = i0 + il;
  const float* si   = ss + i * K;
  const float* arow = A + ((size_t)b * N + i) * N;
  float wsum = 0.f, ecnt = 0.f;
  for (int j = jl; j < N; j += 16) {
    const float* sj = ss + j * K;
    float p = 0.f;
    for (int k = 0; k < K; ++k) p += fabsf(si[k] - sj[k]);
    float a = arow[j];
    wsum += a * p;
    ecnt += (a != 0.f) ? 1.f : 0.f;
  }
  red[2 * threadIdx.x]     = wsum;
  red[2 * threadIdx.x + 1] = ecnt;
  __syncthreads();
  if (threadIdx.x == 0) {
    float w = 0.f, e = 0.f;
    for (int t = 0; t < 256; ++t) { w += red[2 * t]; e += red[2 * t + 1]; }
    int wg = b * gridDim.x + blockIdx.x;
    part[2 * wg] = w; part[2 * wg + 1] = e;
  }
}

// idx-th largest of each cluster column (idx = N/K+1 == 3 here, idx <= 8)
__global__ void k_med(const float* __restrict__ S, float* __restrict__ med,
                      int N, int K, int idx) {
  int t = blockIdx.x * blockDim.x + threadIdx.x;
  if (t >= BATCH * K) return;
  int b = t / K, c = t % K;
  float top[8];
  for (int q = 0; q < idx; ++q) top[q] = -3.4e38f;
  for (int n = 0; n < N; ++n) {
    float v = S[((size_t)b * N + n) * K + c];
    if (v > top[idx - 1]) {
      int q = idx - 1;
      while (q > 0 && v > top[q - 1]) { top[q] = top[q - 1]; --q; }
      top[q] = v;
    }
  }
  med[t] = top[idx - 1];
}

__global__ void k_bal(const float* __restrict__ S, const float* __restrict__ med,
                      float* __restrict__ balp, int N, int K, float kf) {
  __shared__ float red[256];
  int b = blockIdx.x;
  float acc = 0.f;
  for (int t = threadIdx.x; t < N * K; t += 256) {
    int c = t % K;
    float d = S[(size_t)b * N * K + t] - med[b * K + c];
    acc += (d >= 0.f) ? (kf - 1.f) * d : -d;
  }
  red[threadIdx.x] = acc;
  __syncthreads();
  for (int off = 128; off > 0; off >>= 1) {
    if (threadIdx.x < off) red[threadIdx.x] += red[threadIdx.x + off];
    __syncthreads();
  }
  if (threadIdx.x == 0) balp[b] = red[0];
}

// fixed-order scalar reduction -> aux contribution of this level
__global__ void k_aux(const float* __restrict__ part, int ntiles,
                      const float* __restrict__ balp, float* __restrict__ slot,
                      int N, int K) {
  if (threadIdx.x != 0 || blockIdx.x != 0) return;
  float tv = 0.f, bal = 0.f;
  for (int b = 0; b < BATCH; ++b) {
    float w = 0.f, e = 0.f;
    for (int t = 0; t < ntiles; ++t) {
      w += part[2 * (b * ntiles + t)];
      e += part[2 * (b * ntiles + t) + 1];
    }
    tv += w / (2.f * e);
    float nk = (float)N * ((float)K - 1.f);
    bal += (nk - balp[b]) / nk;
  }
  tv /= (float)BATCH; bal /= (float)BATCH;
  *slot = (0.5f / (15.f * 3.f)) * (tv + bal);
}

// SX[b] = S[b]^T @ X[b]   (k x 32)     grid: BATCH*(K/16)*2 waves
__global__ void k_sx(const float* __restrict__ S, const float* __restrict__ X,
                     float* __restrict__ SX, int N, int K) {
  int tile = blockIdx.x;
  int f0 = (tile & 1) << 4; tile >>= 1;
  int nc = K >> 4;
  int c0 = (tile % nc) << 4;
  int b  = tile / nc;
  v8f c = {};
  c = gemm_tile_tn(S + (size_t)b * N * K + c0, K,
                   X + (size_t)b * N * 32 + f0, 32, N, c);
  store_tile(SX + ((size_t)b * K + c0) * 32 + f0, 32, c);
}

// T[b] = A[b] @ S[b]   (N x k)         grid: BATCH*(N/16)*(K/16) waves
__global__ void k_as(const float* __restrict__ A, const float* __restrict__ S,
                     float* __restrict__ T, int N, int K) {
  int tile = blockIdx.x;
  int nc = K >> 4;
  int c0 = (tile % nc) << 4; tile /= nc;
  int nr = N >> 4;
  int i0 = (tile % nr) << 4;
  int b  = tile / nr;
  v8f c = {};
  c = gemm_tile_nn(A + ((size_t)b * N + i0) * N, N,
                   S + (size_t)b * N * K + c0, K, N, c);
  store_tile(T + ((size_t)b * N + i0) * K + c0, K, c);
}

// AN[b] = S[b]^T @ T[b]  (k x k)       grid: BATCH*(K/16)^2 waves
__global__ void k_sts(const float* __restrict__ S, const float* __restrict__ T,
                      float* __restrict__ AN, int N, int K) {
  int tile = blockIdx.x;
  int nc = K >> 4;
  int c1 = (tile % nc) << 4; tile /= nc;
  int c0 = (tile % nc) << 4;
  int b  = tile / nc;
  v8f c = {};
  c = gemm_tile_tn(S + (size_t)b * N * K + c0, K,
                   T + (size_t)b * N * K + c1, K, N, c);
  store_tile(AN + ((size_t)b * K + c0) * K + c1, K, c);
}

__global__ void k_final(const float* __restrict__ X, const float* __restrict__ aux,
                        float* __restrict__ out, int N) {
  int t = blockIdx.x * blockDim.x + threadIdx.x;
  if (t < BATCH * 32) {
    int b = t / 32, f = t % 32;
    float s = 0.f;
    for (int n = 0; n < N; ++n) s += X[((size_t)b * N + n) * 32 + f];
    out[t] = s / (float)N;
  }
  if (t == 0) out[BATCH * 32] = aux[0] + aux[1] + aux[2];
}

// ---------------------------------------------------------------------------
extern "C" void kernel_launch(void* const* d_in, const int* in_sizes, int n_in,
                              void* d_out, int out_size, void* d_ws, size_t ws_size,
                              hipStream_t stream) {
  (void)in_sizes; (void)n_in; (void)out_size; (void)ws_size;
  const float* x_in = (const float*)d_in[0];
  const float* a_in = (const float*)d_in[1];
  const float* cw[4] = {(const float*)d_in[2], (const float*)d_in[4],
                        (const float*)d_in[6], (const float*)d_in[8]};
  const float* cb[4] = {(const float*)d_in[3], (const float*)d_in[5],
                        (const float*)d_in[7], (const float*)d_in[9]};
  const float* pw[3][3];
  const float* pb[3][3];
  for (int p = 0; p < 3; ++p)
    for (int l = 0; l < 3; ++l) {
      pw[p][l] = (const float*)d_in[10 + p * 6 + l * 2];
      pb[p][l] = (const float*)d_in[10 + p * 6 + l * 2 + 1];
    }

  float* ws   = (float*)d_ws;
  float* xw   = ws; ws += BATCH * 256 * 32;
  float* Xc   = ws; ws += BATCH * 256 * 32;   // conv output
  float* Xp   = ws; ws += BATCH * 256 * 32;   // pooled X
  float* AbA  = ws; ws += BATCH * 256 * 256;  // pooled A ping
  float* AbB  = ws; ws += BATCH * 256 * 256;  // pooled A pong
  float* G    = ws; ws += BATCH * 256 * 256;  // gamma
  float* deg  = ws; ws += BATCH * 256;
  float* S    = ws; ws += BATCH * 256 * 128;  // assignments
  float* H0   = ws; ws += BATCH * 256 * 64;
  float* H1   = ws; ws += BATCH * 256 * 96;
  float* T    = ws; ws += BATCH * 256 * 128;  // logits, then A@S
  float* med  = ws; ws += BATCH * 128;
  float* part = ws; ws += 1024;
  float* balp = ws; ws += 16;
  float* auxs = ws; ws += 4;

  const int NS[4] = {256, 128, 64, 32};
  const int D1[3] = {64, 42, 32};
  const int D2[3] = {96, 53, 32};

  const float* Xin  = x_in;
  const float* Ain  = a_in;
  float*       Aout = AbA;

  for (int lvl = 0; lvl < 4; ++lvl) {
    const int N  = NS[lvl];
    const int nt = N / 16;
    // ---- GTV conv ----
    k_gemm_xw<<<BATCH * nt * 2, 32, 0, stream>>>(Xin, cw[lvl], xw, N, 32, 32);
    k_gamma<<<dim3(nt, BATCH), 256, (size_t)(N * 32 + 256) * sizeof(float),
              stream>>>(xw, Ain, G, deg, N);
    k_conv_out<<<BATCH * nt * 2, 32, 0, stream>>>(xw, G, deg, cb[lvl], Xc, N);
    // ---- ACC pooling ----
    if (lvl < 3) {
      const int K = NS[lvl + 1];
      const int d1 = D1[lvl], d2 = D2[lvl];
      const int rows = BATCH * N;
      if (lvl != 1) {
        // dims all multiples of 16 -> WMMA dense with fused bias+squareplus
        k_dense_wmma<<<(rows / 16) * (d1 / 16), 32, 0, stream>>>(Xc, pw[lvl][0], pb[lvl][0], H0, rows, 32, d1, 1);
        k_dense_wmma<<<(rows / 16) * (d2 / 16), 32, 0, stream>>>(H0, pw[lvl][1], pb[lvl][1], H1, rows, d1, d2, 1);
        k_dense_wmma<<<(rows / 16) * (K  / 16), 32, 0, stream>>>(H1, pw[lvl][2], pb[lvl][2], T, rows, d2, K, 0);
      } else {
        k_dense<<<(rows * d1 + 255) / 256, 256, 0, stream>>>(Xc, pw[lvl][0], pb[lvl][0], H0, rows, 32, d1, 1);
        k_dense<<<(rows * d2 + 255) / 256, 256, 0, stream>>>(H0, pw[lvl][1], pb[lvl][1], H1, rows, d1, d2, 1);
        k_dense<<<(rows * K + 255) / 256, 256, 0, stream>>>(H1, pw[lvl][2], pb[lvl][2], T, rows, d2, K, 0);
      }
      k_softmax<<<(rows + 255) / 256, 256, 0, stream>>>(T, S, rows, K);
      k_tv<<<dim3(nt, BATCH), 256, (size_t)(N * K + 512) * sizeof(float),
             stream>>>(S, Ain, part, N, K);
      k_med<<<(BATCH * K + 255) / 256, 256, 0, stream>>>(S, med, N, K, N / K + 1);
      k_bal<<<BATCH, 256, 0, stream>>>(S, med, balp, N, K, (float)K);
      k_aux<<<1, 1, 0, stream>>>(part, nt, balp, auxs + lvl, N, K);
      k_sx<<<BATCH * (K / 16) * 2, 32, 0, stream>>>(S, Xc, Xp, N, K);
      k_as<<<BATCH * nt * (K / 16), 32, 0, stream>>>(Ain, S, T, N, K);
      k_sts<<<BATCH * (K / 16) * (K / 16), 32, 0, stream>>>(S, T, Aout, N, K);
      Xin = Xp;
      Ain = Aout;
      Aout = (Aout == AbA) ? AbB : AbA;
    }
  }
  k_final<<<1, 256, 0, stream>>>(Xc, auxs, (float*)d_out, NS[3]);
}